// HeteroGNNModel_23888608100644
// MI455X (gfx1250) — compile-verified
//
#include <hip/hip_runtime.h>
#include <hip/hip_bf16.h>

typedef __attribute__((ext_vector_type(2))) float v2f;
typedef __attribute__((ext_vector_type(8))) float v8f;

#define NPAPER   100000
#define NAUTHOR  50000
#define DIM      128
#define NEDGE    500000
#define NCLS     16
#define NLAYER   2
#define DD       (DIM * DIM)

// ---------------------------------------------------------------- utilities

__global__ void zero_buf(float* __restrict__ p, int n) {
    int i = blockIdx.x * blockDim.x + threadIdx.x;
    if (i < n) p[i] = 0.0f;
}

// One wave per edge: gather a 128-float source row, atomic-add into dst row,
// lane 0 bumps the count.
__global__ void scatter_mean(const float* __restrict__ xsrc,
                             const int* __restrict__ ei, int E,
                             float* __restrict__ acc, float* __restrict__ cnt) {
    int e    = (blockIdx.x * blockDim.x + threadIdx.x) >> 5;
    int lane = threadIdx.x & 31;
    if (e >= E) return;
    int src = ei[e];
    int dst = ei[E + e];
    float4 v = ((const float4*)(xsrc + (size_t)src * DIM))[lane];
    float* a = acc + (size_t)dst * DIM + lane * 4;
    atomicAdd(a + 0, v.x);
    atomicAdd(a + 1, v.y);
    atomicAdd(a + 2, v.z);
    atomicAdd(a + 3, v.w);
    if (lane == 0) atomicAdd(cnt + dst, 1.0f);
}

__global__ void finalize_mean(float* __restrict__ acc,
                              const float* __restrict__ cnt, int n) {
    int i = blockIdx.x * blockDim.x + threadIdx.x;  // over n*DIM
    if (i >= n * DIM) return;
    float c = cnt[i >> 7];
    acc[i] *= 1.0f / fmaxf(c, 1.0f);
}

// ------------------------------------------------------------- main GEMM
// out[m,n] = (bias?bias[n]:0) + (addC?addC[m,n]:0)
//          + sum_k A1[m,k]*W1[n,k] + (A2? sum_k A2[m,k]*W2[n,k] : 0)
// 256 threads = 8 waves; each wave computes a 16x128 output strip via
// v_wmma_f32_16x16x4_f32. Weights staged in LDS as K-pair-interleaved
// transposed float2s: sP[(k>>1)*128 + n] = {W[n][k], W[n][k+1]} so each
// B fragment is one contiguous ds_load_b64 per lane.
__global__ void gemm128(const float* __restrict__ A1, const float* __restrict__ W1,
                        const float* __restrict__ A2, const float* __restrict__ W2,
                        const float* __restrict__ bias, const float* __restrict__ addC,
                        float* __restrict__ out, int M) {
    __shared__ float2 sP[64 * DIM];  // 64KB

    const int tid  = threadIdx.x;
    const int lane = tid & 31;
    const int wave = tid >> 5;

    const int rowBase = blockIdx.x * 128 + wave * 16;
    const int nSub  = lane & 15;           // N within 16-tile (B/C layout)
    const int mA    = lane & 15;           // M within tile (A layout)
    const int kPair = (lane >> 4) << 1;    // K offset 0 or 2 (A/B layout)
    const int mOff  = (lane >> 4) << 3;    // M offset 0 or 8 (C/D layout)

    const int rClamp = (rowBase + mA) < M ? (rowBase + mA) : (M - 1);

    // prefetch pass-2 operands while we stage + compute pass 1
    if (A2) {
        __builtin_prefetch(A2 + (size_t)rClamp * DIM, 0, 3);
        __builtin_prefetch(W2 + tid * 16, 0, 3);
    }

    // stage W1 pairs (transpose: global float2 i = {n, kh} -> sP[kh*128 + n])
    {
        const float2* Wp = (const float2*)W1;
        for (int i = tid; i < DIM * 64; i += 256) {
            int n = i >> 6, kh = i & 63;
            sP[kh * DIM + n] = Wp[i];
        }
    }
    __syncthreads();

    // accumulator init: bias + optional C
    v8f acc[8];
#pragma unroll
    for (int t = 0; t < 8; ++t) {
        float bv = bias ? bias[t * 16 + nSub] : 0.0f;
#pragma unroll
        for (int r = 0; r < 8; ++r) {
            int row = rowBase + mOff + r;
            int rr  = row < M ? row : (M - 1);
            float v = bv;
            if (addC) v += addC[(size_t)rr * DIM + t * 16 + nSub];
            acc[t][r] = v;
        }
    }

    // pass 1: A1 x W1^T
    {
        const float* Arow = A1 + (size_t)rClamp * DIM;
        for (int kk = 0; kk < DIM; kk += 4) {
            v2f a;
            a.x = Arow[kk + kPair];
            a.y = Arow[kk + kPair + 1];
            const int kh = ((kk + kPair) >> 1) * DIM;
#pragma unroll
            for (int t = 0; t < 8; ++t) {
                float2 bb = sP[kh + t * 16 + nSub];
                v2f b;
                b.x = bb.x;
                b.y = bb.y;
                acc[t] = __builtin_amdgcn_wmma_f32_16x16x4_f32(
                    false, a, false, b, (short)0, acc[t], false, false);
            }
        }
    }

    // pass 2: A2 x W2^T (optional)
    if (A2) {
        __syncthreads();
        {
            const float2* Wp = (const float2*)W2;
            for (int i = tid; i < DIM * 64; i += 256) {
                int n = i >> 6, kh = i & 63;
                sP[kh * DIM + n] = Wp[i];
            }
        }
        __syncthreads();
        const float* Arow = A2 + (size_t)rClamp * DIM;
        for (int kk = 0; kk < DIM; kk += 4) {
            v2f a;
            a.x = Arow[kk + kPair];
            a.y = Arow[kk + kPair + 1];
            const int kh = ((kk + kPair) >> 1) * DIM;
#pragma unroll
            for (int t = 0; t < 8; ++t) {
                float2 bb = sP[kh + t * 16 + nSub];
                v2f b;
                b.x = bb.x;
                b.y = bb.y;
                acc[t] = __builtin_amdgcn_wmma_f32_16x16x4_f32(
                    false, a, false, b, (short)0, acc[t], false, false);
            }
        }
    }

    // store (C/D layout: vgpr r -> M = mOff + r, N = t*16 + nSub)
#pragma unroll
    for (int t = 0; t < 8; ++t)
#pragma unroll
        for (int r = 0; r < 8; ++r) {
            int row = rowBase + mOff + r;
            if (row < M) out[(size_t)row * DIM + t * 16 + nSub] = acc[t][r];
        }
}

// ---------------------------------------------------------- LayerNorm+ReLU
__global__ void ln_relu(const float* __restrict__ in,
                        const float* __restrict__ g, const float* __restrict__ b,
                        float* __restrict__ out, int n) {
    int row  = (blockIdx.x * blockDim.x + threadIdx.x) >> 5;
    int lane = threadIdx.x & 31;
    if (row >= n) return;
    float4 v = ((const float4*)(in + (size_t)row * DIM))[lane];
    float s = v.x + v.y + v.z + v.w;
    for (int off = 16; off; off >>= 1) s += __shfl_xor(s, off, 32);
    float m = s * (1.0f / DIM);
    float4 d = make_float4(v.x - m, v.y - m, v.z - m, v.w - m);
    float q = d.x * d.x + d.y * d.y + d.z * d.z + d.w * d.w;
    for (int off = 16; off; off >>= 1) q += __shfl_xor(q, off, 32);
    float inv = rsqrtf(q * (1.0f / DIM) + 1e-5f);
    float4 gg = ((const float4*)g)[lane];
    float4 bb = ((const float4*)b)[lane];
    float4 o;
    o.x = fmaxf(d.x * inv * gg.x + bb.x, 0.0f);
    o.y = fmaxf(d.y * inv * gg.y + bb.y, 0.0f);
    o.z = fmaxf(d.z * inv * gg.z + bb.z, 0.0f);
    o.w = fmaxf(d.w * inv * gg.w + bb.w, 0.0f);
    ((float4*)(out + (size_t)row * DIM))[lane] = o;
}

// ------------------------------------------------------------- head GEMM
// out[m,c] = sum_k X[m,k] * Wh[k,c] + bh[c]   (Wh is [128,16], K-major)
// LDS holds K-pair-interleaved Wh: sPh[kh*16 + c] = {Wh[2kh][c], Wh[2kh+1][c]}
__global__ void head_gemm(const float* __restrict__ X, const float* __restrict__ Wh,
                          const float* __restrict__ bh, float* __restrict__ out, int M) {
    __shared__ float2 sPh[64 * NCLS];
    int tid = threadIdx.x;
    for (int i = tid; i < 64 * NCLS; i += 256) {
        int kh = i >> 4, c = i & 15;
        sPh[i] = make_float2(Wh[(2 * kh) * NCLS + c], Wh[(2 * kh + 1) * NCLS + c]);
    }
    __syncthreads();

    int lane = tid & 31, wave = tid >> 5;
    int rowBase = blockIdx.x * 128 + wave * 16;
    int nSub  = lane & 15;
    int mA    = lane & 15;
    int kPair = (lane >> 4) << 1;
    int mOff  = (lane >> 4) << 3;

    v8f acc;
    float bv = bh[nSub];
#pragma unroll
    for (int r = 0; r < 8; ++r) acc[r] = bv;

    int rClamp = (rowBase + mA) < M ? (rowBase + mA) : (M - 1);
    const float* Arow = X + (size_t)rClamp * DIM;
    for (int kk = 0; kk < DIM; kk += 4) {
        v2f a;
        a.x = Arow[kk + kPair];
        a.y = Arow[kk + kPair + 1];
        float2 bb = sPh[((kk + kPair) >> 1) * NCLS + nSub];
        v2f b;
        b.x = bb.x;
        b.y = bb.y;
        acc = __builtin_amdgcn_wmma_f32_16x16x4_f32(
            false, a, false, b, (short)0, acc, false, false);
    }
#pragma unroll
    for (int r = 0; r < 8; ++r) {
        int row = rowBase + mOff + r;
        if (row < M) out[(size_t)row * NCLS + nSub] = acc[r];
    }
}

// ------------------------------------------------------------------ launch

static inline int ceil_div(int a, int b) { return (a + b - 1) / b; }

extern "C" void kernel_launch(void* const* d_in, const int* in_sizes, int n_in,
                              void* d_out, int out_size, void* d_ws, size_t ws_size,
                              hipStream_t stream) {
    const float* x_paper   = (const float*)d_in[0];
    const float* x_author  = (const float*)d_in[1];
    const int*   ei_cites  = (const int*)d_in[2];
    const int*   ei_writes = (const int*)d_in[3];
    const int*   ei_rev    = (const int*)d_in[4];
    const float* Wl        = (const float*)d_in[5];   // [L,3,D,D]
    const float* Wr        = (const float*)d_in[6];   // [L,3,D,D]
    const float* bl        = (const float*)d_in[7];   // [L,3,D]
    const float* lin_p     = (const float*)d_in[8];   // [D,D]
    const float* lin_a     = (const float*)d_in[9];   // [D,D]
    const float* ln_s      = (const float*)d_in[10];  // [L,2,D]
    const float* ln_b      = (const float*)d_in[11];  // [L,2,D]
    const float* Wh        = (const float*)d_in[12];  // [D,C]
    const float* bh        = (const float*)d_in[13];  // [C]

    float* ws   = (float*)d_ws;
    float* xp   = ws;                                   // NP*D
    float* xa   = xp + (size_t)NPAPER * DIM;            // NA*D
    float* pnew = xa + (size_t)NAUTHOR * DIM;           // NP*D
    float* anew = pnew + (size_t)NPAPER * DIM;          // NA*D
    float* mean = anew + (size_t)NAUTHOR * DIM;         // NP*D
    float* cnt  = mean + (size_t)NPAPER * DIM;          // NP

    const dim3 blk(256);
    const int scatter_blocks = ceil_div(NEDGE * 32, 256);

    for (int l = 0; l < NLAYER; ++l) {
        const float* xpl = (l == 0) ? x_paper : xp;
        const float* xal = (l == 0) ? x_author : xa;

        // ---- paper <- cites (paper -> paper)
        zero_buf<<<ceil_div(NPAPER * DIM, 256), blk, 0, stream>>>(mean, NPAPER * DIM);
        zero_buf<<<ceil_div(NPAPER, 256), blk, 0, stream>>>(cnt, NPAPER);
        scatter_mean<<<scatter_blocks, blk, 0, stream>>>(xpl, ei_cites, NEDGE, mean, cnt);
        finalize_mean<<<ceil_div(NPAPER * DIM, 256), blk, 0, stream>>>(mean, cnt, NPAPER);
        gemm128<<<ceil_div(NPAPER, 128), blk, 0, stream>>>(
            mean, Wl + (size_t)(l * 3 + 0) * DD, xpl, Wr + (size_t)(l * 3 + 0) * DD,
            bl + (size_t)(l * 3 + 0) * DIM, nullptr, pnew, NPAPER);

        // ---- paper <- writes (author -> paper), accumulate into pnew
        zero_buf<<<ceil_div(NPAPER * DIM, 256), blk, 0, stream>>>(mean, NPAPER * DIM);
        zero_buf<<<ceil_div(NPAPER, 256), blk, 0, stream>>>(cnt, NPAPER);
        scatter_mean<<<scatter_blocks, blk, 0, stream>>>(xal, ei_writes, NEDGE, mean, cnt);
        finalize_mean<<<ceil_div(NPAPER * DIM, 256), blk, 0, stream>>>(mean, cnt, NPAPER);
        gemm128<<<ceil_div(NPAPER, 128), blk, 0, stream>>>(
            mean, Wl + (size_t)(l * 3 + 1) * DD, xpl, Wr + (size_t)(l * 3 + 1) * DD,
            bl + (size_t)(l * 3 + 1) * DIM, pnew, pnew, NPAPER);

        // ---- author <- rev (paper -> author)
        zero_buf<<<ceil_div(NAUTHOR * DIM, 256), blk, 0, stream>>>(mean, NAUTHOR * DIM);
        zero_buf<<<ceil_div(NAUTHOR, 256), blk, 0, stream>>>(cnt, NAUTHOR);
        scatter_mean<<<scatter_blocks, blk, 0, stream>>>(xpl, ei_rev, NEDGE, mean, cnt);
        finalize_mean<<<ceil_div(NAUTHOR * DIM, 256), blk, 0, stream>>>(mean, cnt, NAUTHOR);
        gemm128<<<ceil_div(NAUTHOR, 128), blk, 0, stream>>>(
            mean, Wl + (size_t)(l * 3 + 2) * DD, xal, Wr + (size_t)(l * 3 + 2) * DD,
            bl + (size_t)(l * 3 + 2) * DIM, nullptr, anew, NAUTHOR);

        if (l < NLAYER - 1) {
            // residual linear: t = pnew + pnew @ lin^T, then LN+ReLU
            gemm128<<<ceil_div(NPAPER, 128), blk, 0, stream>>>(
                pnew, lin_p, nullptr, nullptr, nullptr, pnew, mean, NPAPER);
            ln_relu<<<ceil_div(NPAPER * 32, 256), blk, 0, stream>>>(
                mean, ln_s + (size_t)(l * 2 + 0) * DIM, ln_b + (size_t)(l * 2 + 0) * DIM,
                xp, NPAPER);
            gemm128<<<ceil_div(NAUTHOR, 128), blk, 0, stream>>>(
                anew, lin_a, nullptr, nullptr, nullptr, anew, mean, NAUTHOR);
            ln_relu<<<ceil_div(NAUTHOR * 32, 256), blk, 0, stream>>>(
                mean, ln_s + (size_t)(l * 2 + 1) * DIM, ln_b + (size_t)(l * 2 + 1) * DIM,
                xa, NAUTHOR);
        } else {
            ln_relu<<<ceil_div(NPAPER * 32, 256), blk, 0, stream>>>(
                pnew, ln_s + (size_t)(l * 2 + 0) * DIM, ln_b + (size_t)(l * 2 + 0) * DIM,
                xp, NPAPER);
            ln_relu<<<ceil_div(NAUTHOR * 32, 256), blk, 0, stream>>>(
                anew, ln_s + (size_t)(l * 2 + 1) * DIM, ln_b + (size_t)(l * 2 + 1) * DIM,
                xa, NAUTHOR);
        }
    }

    // head on paper nodes
    head_gemm<<<ceil_div(NPAPER, 128), blk, 0, stream>>>(xp, Wh, bh, (float*)d_out, NPAPER);
}